// Memory_54546084660145
// MI455X (gfx1250) — compile-verified
//
#include <hip/hip_runtime.h>
#include <cstdint>
#include <cstddef>

// Problem constants (from setup_inputs): B=64, N=16384, M=64
constexpr int  Bc = 64;
constexpr int  Nc = 16384;
constexpr int  Mc = 64;
constexpr float EPSF = 1e-16f;

typedef float v2f __attribute__((ext_vector_type(2)));
typedef float v4f __attribute__((ext_vector_type(4)));
typedef float v8f __attribute__((ext_vector_type(8)));

#if defined(__has_builtin)
#  if __has_builtin(__builtin_amdgcn_wmma_f32_16x16x4_f32)
#    define HAVE_WMMA4 1
#  endif
#endif
#ifndef HAVE_WMMA4
#  define HAVE_WMMA4 0
#endif

// ---------------------------------------------------------------------------
// K0: key norms + zero read-output (64 blocks x 64 threads)
// ---------------------------------------------------------------------------
__global__ __launch_bounds__(64) void ntm_prep_kernel(const float* __restrict__ key,
                                                      float* __restrict__ knorm,
                                                      float* __restrict__ out_read) {
    __shared__ float red[64];
    const int b = blockIdx.x, t = threadIdx.x;
    const float k = key[b * Mc + t];
    red[t] = k * k;
    __syncthreads();
    for (int s = 32; s > 0; s >>= 1) {
        if (t < s) red[t] += red[t + s];
        __syncthreads();
    }
    if (t == 0) knorm[b] = sqrtf(red[0]);
    out_read[b * Mc + t] = 0.0f;   // re-initialized every launch (deterministic)
}

// ---------------------------------------------------------------------------
// K1: cosine-similarity pass. One wave32 = 16 memory rows.
//     dots via V_WMMA_F32_16X16X4_F32 (K accumulated over 16 chunks of 4),
//     row norms via VALU on the same loaded registers.
//     memory is a 256MB single-use stream -> non-temporal loads.
// A layout (16x4 f32): lane L holds row (L&15), K = k0 + 2*(L>>4) + {0,1}
// B layout (4x16 f32): same K split per lane half; all 16 columns = key slice
// D layout: VGPR v -> M = v + 8*(L>>4), column = L&15 (all columns equal)
// ---------------------------------------------------------------------------
__global__ __launch_bounds__(256) void ntm_sim_kernel(const float* __restrict__ mem,
                                                      const float* __restrict__ key,
                                                      const float* __restrict__ knorm,
                                                      float* __restrict__ sim) {
    const int lane = threadIdx.x & 31;
    const int wave = threadIdx.x >> 5;
    const int half = lane >> 4;
    const int r    = lane & 15;

    const long   chunk    = (long)blockIdx.x * 8 + wave;   // 16-row chunk
    const size_t row_base = (size_t)chunk * 16;            // global row index
    const int    b        = (int)(chunk >> 10);            // 1024 chunks / batch

    const float* rowp = mem + (row_base + (size_t)r) * Mc;
    const float* keyp = key + b * Mc;
    const float  kn   = knorm[b];

    v8f acc;
    #pragma unroll
    for (int i = 0; i < 8; ++i) acc[i] = 0.0f;
    float nrm  = 0.0f;
#if !HAVE_WMMA4
    float dotp = 0.0f;
#endif

    #pragma unroll
    for (int k0 = 0; k0 < Mc; k0 += 4) {
        const int off = k0 + 2 * half;
        const v2f a  = __builtin_nontemporal_load((const v2f*)(rowp + off));
        const v2f kv = *(const v2f*)(keyp + off);   // key: tiny, keep cached
        nrm += a.x * a.x + a.y * a.y;
#if HAVE_WMMA4
        acc = __builtin_amdgcn_wmma_f32_16x16x4_f32(
            /*neg_a=*/false, a, /*neg_b=*/false, kv,
            /*c_mod=*/(short)0, acc, /*reuse_a=*/false, /*reuse_b=*/false);
#else
        dotp += a.x * kv.x + a.y * kv.y;
#endif
    }

    // combine the two K-halves of the row norm (wave32 xor-16 shuffle)
    const float nfull = nrm + __shfl_xor(nrm, 16, 32);

#if HAVE_WMMA4
    #pragma unroll
    for (int v = 0; v < 8; ++v) {
        const int   row = half * 8 + v;              // D-matrix M index
        const float nr  = __shfl(nfull, row, 32);    // norm for that row
        const float s   = acc[v] / fmaxf(sqrtf(nr) * kn, EPSF);
        if (r == 0) sim[row_base + row] = s;         // lanes 0 and 16 store
    }
#else
    const float dfull = dotp + __shfl_xor(dotp, 16, 32);
    if (half == 0) sim[row_base + r] = dfull / fmaxf(sqrtf(nfull) * kn, EPSF);
#endif
}

// ---------------------------------------------------------------------------
// K2: per-batch addressing: softmax -> interpolate -> circular shift -> sharpen
//     One block (1024 threads) per batch; 16 contiguous elements per thread.
//     All data here is small (4MB-class) and L2-resident -> regular RT hints.
// ---------------------------------------------------------------------------
__device__ __forceinline__ float blockReduceMax(float v, float* red) {
    const int t = threadIdx.x;
    red[t] = v; __syncthreads();
    for (int s = 512; s > 0; s >>= 1) {
        if (t < s) red[t] = fmaxf(red[t], red[t + s]);
        __syncthreads();
    }
    const float r = red[0]; __syncthreads();
    return r;
}
__device__ __forceinline__ float blockReduceSum(float v, float* red) {
    const int t = threadIdx.x;
    red[t] = v; __syncthreads();
    for (int s = 512; s > 0; s >>= 1) {
        if (t < s) red[t] += red[t + s];
        __syncthreads();
    }
    const float r = red[0]; __syncthreads();
    return r;
}

__global__ __launch_bounds__(1024) void ntm_addr_kernel(const float* sim,      // aliases wgbuf
                                                        const float* __restrict__ beta_,
                                                        const float* __restrict__ gate_,
                                                        const float* __restrict__ shiftw,
                                                        const float* __restrict__ sharp_,
                                                        const float* __restrict__ prevw,
                                                        float* wgbuf,           // aliases sim
                                                        float* __restrict__ out_w) {
    __shared__ float red[1024];
    const int b = blockIdx.x, t = threadIdx.x;
    const size_t base = (size_t)b * Nc;
    const int i0 = t * 16;

    const float beta = beta_[b];
    const float g    = gate_[b];
    const float s0   = shiftw[b * 3 + 0];
    const float s1   = shiftw[b * 3 + 1];
    const float s2   = shiftw[b * 3 + 2];
    const float p    = sharp_[b];

    // load 16 contiguous sim values (vectorized)
    float x[16];
    #pragma unroll
    for (int q = 0; q < 4; ++q) {
        const v4f v = ((const v4f*)(sim + base + i0))[q];
        x[4*q+0] = v.x; x[4*q+1] = v.y; x[4*q+2] = v.z; x[4*q+3] = v.w;
    }

    // softmax over beta*sim
    float mx = -3.4e38f;
    #pragma unroll
    for (int j = 0; j < 16; ++j) mx = fmaxf(mx, beta * x[j]);
    mx = blockReduceMax(mx, red);

    float e[16]; float lsum = 0.0f;
    #pragma unroll
    for (int j = 0; j < 16; ++j) { e[j] = expf(beta * x[j] - mx); lsum += e[j]; }
    const float Z = blockReduceSum(lsum, red);
    const float invZ = 1.0f / Z;

    // interpolate with previous weights -> wg, write to scratch for neighbors
    #pragma unroll
    for (int q = 0; q < 4; ++q) {
        const v4f pv = ((const v4f*)(prevw + base + i0))[q];
        x[4*q+0] = g * e[4*q+0] * invZ + (1.0f - g) * pv.x;
        x[4*q+1] = g * e[4*q+1] * invZ + (1.0f - g) * pv.y;
        x[4*q+2] = g * e[4*q+2] * invZ + (1.0f - g) * pv.z;
        x[4*q+3] = g * e[4*q+3] * invZ + (1.0f - g) * pv.w;
        v4f o; o.x = x[4*q+0]; o.y = x[4*q+1]; o.z = x[4*q+2]; o.w = x[4*q+3];
        ((v4f*)(wgbuf + base + i0))[q] = o;
    }
    __threadfence();
    __syncthreads();

    // circular shift: ws[i] = s0*wg[i-1] + s1*wg[i] + s2*wg[i+1]
    const float wl = wgbuf[base + (size_t)((i0 + Nc - 1) % Nc)];
    const float wr = wgbuf[base + (size_t)((i0 + 16) % Nc)];
    float ssum = 0.0f;
    #pragma unroll
    for (int j = 0; j < 16; ++j) {
        const float left  = (j > 0)  ? x[j - 1] : wl;
        const float right = (j < 15) ? x[j + 1] : wr;
        const float ws = s0 * left + s1 * x[j] + s2 * right;
        e[j] = powf(ws, p);                 // reuse e[] for sharpened weights
        ssum += e[j];
    }
    const float S = blockReduceSum(ssum, red);
    const float inv = 1.0f / (S + EPSF);

    #pragma unroll
    for (int q = 0; q < 4; ++q) {
        v4f o;
        o.x = e[4*q+0] * inv; o.y = e[4*q+1] * inv;
        o.z = e[4*q+2] * inv; o.w = e[4*q+3] * inv;
        ((v4f*)(out_w + base + i0))[q] = o;
    }
}

// ---------------------------------------------------------------------------
// K3: memory update + read reduction. 64 rows per block (4 x 16 rows),
//     16 lanes per row, v4f per lane. Block-level LDS reduce then
//     64 atomicAdds per block into out_read.
//     memory re-read (256MB) + new_memory write (256MB) are pure streams that
//     can never be reused -> non-temporal loads AND stores (avoid L2 thrash /
//     write-allocate). w / erase / add stay RT (small, L2-resident from K2).
// ---------------------------------------------------------------------------
__global__ __launch_bounds__(256) void ntm_update_kernel(const float* __restrict__ mem,
                                                         const float* __restrict__ w,
                                                         const float* __restrict__ ev,
                                                         const float* __restrict__ av,
                                                         float* __restrict__ out_nm,
                                                         float* __restrict__ out_read) {
    __shared__ v4f red[256];
    const int t   = threadIdx.x;
    const int g16 = t >> 4;      // row group 0..15
    const int q   = t & 15;      // m-quad 0..15
    const int b   = blockIdx.x >> 8;                // 256 blocks per batch
    const int nb  = (blockIdx.x & 255) * 64;        // first row in block
    const size_t mbase = ((size_t)b * Nc + nb) * Mc;

    const v4f e4 = ((const v4f*)(ev + b * Mc))[q];
    const v4f a4 = ((const v4f*)(av + b * Mc))[q];
    v4f d4; d4.x = a4.x - e4.x; d4.y = a4.y - e4.y;
            d4.z = a4.z - e4.z; d4.w = a4.w - e4.w;

    v4f acc; acc.x = acc.y = acc.z = acc.w = 0.0f;

    #pragma unroll
    for (int it = 0; it < 4; ++it) {
        const int row = it * 16 + g16;
        const float wv = w[(size_t)b * Nc + nb + row];
        const v4f m4 =
            __builtin_nontemporal_load((const v4f*)(mem + mbase + (size_t)row * Mc) + q);
        v4f o;
        o.x = m4.x * (1.0f + wv * d4.x);
        o.y = m4.y * (1.0f + wv * d4.y);
        o.z = m4.z * (1.0f + wv * d4.z);
        o.w = m4.w * (1.0f + wv * d4.w);
        __builtin_nontemporal_store(o, (v4f*)(out_nm + mbase + (size_t)row * Mc) + q);
        acc.x += wv * m4.x; acc.y += wv * m4.y;
        acc.z += wv * m4.z; acc.w += wv * m4.w;
    }

    red[t] = acc;
    __syncthreads();
    for (int s = 8; s > 0; s >>= 1) {
        if (g16 < s) {
            red[t].x += red[t + s * 16].x; red[t].y += red[t + s * 16].y;
            red[t].z += red[t + s * 16].z; red[t].w += red[t + s * 16].w;
        }
        __syncthreads();
    }
    if (g16 == 0) {
        float* dst = out_read + b * Mc + q * 4;
        atomicAdd(dst + 0, red[t].x);
        atomicAdd(dst + 1, red[t].y);
        atomicAdd(dst + 2, red[t].z);
        atomicAdd(dst + 3, red[t].w);
    }
}

// ---------------------------------------------------------------------------
extern "C" void kernel_launch(void* const* d_in, const int* in_sizes, int n_in,
                              void* d_out, int out_size, void* d_ws, size_t ws_size,
                              hipStream_t stream) {
    const float* mem    = (const float*)d_in[0];
    const float* key    = (const float*)d_in[1];
    const float* beta   = (const float*)d_in[2];
    const float* gate   = (const float*)d_in[3];
    const float* shiftw = (const float*)d_in[4];
    const float* sharp  = (const float*)d_in[5];
    const float* prevw  = (const float*)d_in[6];
    const float* ev     = (const float*)d_in[7];
    const float* av     = (const float*)d_in[8];

    float* out_read = (float*)d_out;                            // B*M
    float* out_nm   = out_read + (size_t)Bc * Mc;               // B*N*M
    float* out_w    = out_nm + (size_t)Bc * Nc * Mc;            // B*N

    float* sim   = (float*)d_ws;                                // B*N floats
    float* knorm = sim + (size_t)Bc * Nc;                       // B floats

    ntm_prep_kernel<<<Bc, 64, 0, stream>>>(key, knorm, out_read);
    ntm_sim_kernel<<<(Bc * Nc) / 128, 256, 0, stream>>>(mem, key, knorm, sim);
    // wg buffer aliases sim (sim fully consumed before wg is written)
    ntm_addr_kernel<<<Bc, 1024, 0, stream>>>(sim, beta, gate, shiftw, sharp,
                                             prevw, sim, out_w);
    ntm_update_kernel<<<(Bc * Nc) / 64, 256, 0, stream>>>(mem, out_w, ev, av,
                                                          out_nm, out_read);
}